// VTSNE_81690277970638
// MI455X (gfx1250) — compile-verified
//
#include <hip/hip_runtime.h>
#include <hip/hip_bf16.h>

#define N_POINTS 16384
#define N_PAIRS  65536

typedef __attribute__((ext_vector_type(2))) float v2f;
typedef __attribute__((ext_vector_type(8))) float v8f;

// ---------------------------------------------------------------------------
// Block reduction helper (deterministic LDS tree, 256 threads)
// ---------------------------------------------------------------------------
__device__ __forceinline__ float block_reduce_256(float v) {
    __shared__ float sm[256];
    __syncthreads();                 // protect against WAR with a previous call
    sm[threadIdx.x] = v;
    __syncthreads();
    #pragma unroll
    for (int o = 128; o > 0; o >>= 1) {
        if ((int)threadIdx.x < o) sm[threadIdx.x] += sm[threadIdx.x + o];
        __syncthreads();
    }
    return sm[0];
}

// ---------------------------------------------------------------------------
// Kernel 1: reparametrize full table, build WMMA-friendly A/B point tables,
//           and reduce the KLD regularizer.
//   Ar[p] = ( x0, x1, x2_p, 1 )      (A-matrix rows, K=0..3)
//   Bc[p] = (-2x0, -2x1, 1, x2_p )   (B-matrix cols, K=0..3)
//   => A_row(m) . B_col(n) = x2_m + x2_n - 2*x_m.x_n = dkl2 (pre-clamp)
// ---------------------------------------------------------------------------
__global__ __launch_bounds__(256)
void vtsne_prep(const float* __restrict__ mu, const float* __restrict__ lv,
                const float* __restrict__ eps,
                float4* __restrict__ Ar, float4* __restrict__ Bc,
                float* __restrict__ kld_partials) {
    const int p = blockIdx.x * 256 + threadIdx.x;   // exact: 64*256 = 16384
    const float mu0 = mu[2 * p], mu1 = mu[2 * p + 1];
    const float lv0 = lv[2 * p], lv1 = lv[2 * p + 1];
    const float e0  = eps[2 * p], e1 = eps[2 * p + 1];

    const float ev0 = __expf(lv0), ev1 = __expf(lv1);
    const float x0 = fmaf(e0, __expf(0.5f * lv0), mu0);
    const float x1 = fmaf(e1, __expf(0.5f * lv1), mu1);
    const float x2 = fmaf(x0, x0, x1 * x1);

    Ar[p] = make_float4(x0, x1, x2, 1.0f);
    Bc[p] = make_float4(-2.0f * x0, -2.0f * x1, 1.0f, x2);

    const float kld = -0.5f * ((1.0f + lv0 - mu0 * mu0 - ev0) +
                               (1.0f + lv1 - mu1 * mu1 - ev1));
    const float r = block_reduce_256(kld);
    if (threadIdx.x == 0) kld_partials[blockIdx.x] = r;
}

// ---------------------------------------------------------------------------
// Kernel 2: all-pairs normalizer via V_WMMA_F32_16X16X4_F32.
// grid = (1024 row-blocks, 8 column slices), 32 threads (one wave32).
// A tile layout (ISA 7.12.2, 32-bit A 16x4): lanes 0-15 hold K=0,1 of row M=lane;
// lanes 16-31 hold K=2,3 of row M=lane-16.  B mirrors with N=lane&15.
// C tile layout: lane<16 -> N=lane, M=vgpr; lane>=16 -> N=lane-16, M=vgpr+8.
// ---------------------------------------------------------------------------
__global__ __launch_bounds__(32)
void vtsne_part_tiles(const float2* __restrict__ A2, const float2* __restrict__ B2,
                      float* __restrict__ part_partials) {
    const int lane = threadIdx.x;          // 0..31
    const int l15  = lane & 15;
    const int hi   = (lane >> 4) & 1;      // 0: K=0,1  1: K=2,3

    // Load A tile once: row-block fixed per workgroup.
    const int row = blockIdx.x * 16 + l15;
    const float2 av = A2[row * 2 + hi];
    const v2f a = { av.x, av.y };

    float s = 0.0f;
    const int nbase = blockIdx.y * 128;    // this slice covers 128 column tiles
    #pragma unroll 2
    for (int t = 0; t < 128; ++t) {
        const int col = (nbase + t) * 16 + l15;
        const float2 bv = B2[col * 2 + hi];
        const v2f b = { bv.x, bv.y };

        v8f c = {};  // fresh accumulator: tile is consumed nonlinearly
        c = __builtin_amdgcn_wmma_f32_16x16x4_f32(
                /*neg_a=*/false, a, /*neg_b=*/false, b,
                /*c_mod=*/(short)0, c, /*reuse_a=*/false, /*reuse_b=*/false);

        #pragma unroll
        for (int v = 0; v < 8; ++v) {
            const float d = fmaxf(c[v], 0.0f);        // dkl2 clamp
            s += __builtin_amdgcn_rcpf(1.0f + d);     // 1/(1+dkl2)
        }
    }

    // wave32 reduction
    #pragma unroll
    for (int o = 16; o > 0; o >>= 1) s += __shfl_xor(s, o, 32);
    if (lane == 0) part_partials[blockIdx.y * 1024 + blockIdx.x] = s;
}

// ---------------------------------------------------------------------------
// Kernel 3: per-pair terms. Accumulate
//   S1 = sum pij*(log pij + log(1+d2))   and   Sp = sum pij
// (log(part) folded in at finalize since qij = num/part)
// ---------------------------------------------------------------------------
__global__ __launch_bounds__(256)
void vtsne_pairs(const float* __restrict__ pij,
                 const int* __restrict__ ii, const int* __restrict__ jj,
                 const float* __restrict__ mu, const float* __restrict__ lv,
                 const float* __restrict__ ei, const float* __restrict__ ej,
                 float* __restrict__ pair_partials) {
    const int t = blockIdx.x * 256 + threadIdx.x;   // exact: 256*256 = 65536
    const int i = ii[t], j = jj[t];

    const float xi0 = fmaf(ei[2 * t],     __expf(0.5f * lv[2 * i]),     mu[2 * i]);
    const float xi1 = fmaf(ei[2 * t + 1], __expf(0.5f * lv[2 * i + 1]), mu[2 * i + 1]);
    const float xj0 = fmaf(ej[2 * t],     __expf(0.5f * lv[2 * j]),     mu[2 * j]);
    const float xj1 = fmaf(ej[2 * t + 1], __expf(0.5f * lv[2 * j + 1]), mu[2 * j + 1]);

    const float d0 = xi0 - xj0, d1 = xi1 - xj1;
    const float d2 = fmaf(d0, d0, d1 * d1);

    const float p  = pij[t];
    const float s1 = p * (__logf(p) + __logf(1.0f + d2));  // -log(num)=log(1+d2)

    const float r1 = block_reduce_256(s1);
    const float rp = block_reduce_256(p);
    if (threadIdx.x == 0) {
        pair_partials[2 * blockIdx.x]     = r1;
        pair_partials[2 * blockIdx.x + 1] = rp;
    }
}

// ---------------------------------------------------------------------------
// Kernel 4: finalize (single block, fixed-order reductions)
// ---------------------------------------------------------------------------
__global__ __launch_bounds__(256)
void vtsne_finalize(const float* __restrict__ part_partials,
                    const float* __restrict__ kld_partials,
                    const float* __restrict__ pair_partials,
                    float* __restrict__ out) {
    float sp = 0.0f, sk = 0.0f, s1 = 0.0f, spp = 0.0f;
    for (int t = threadIdx.x; t < 8192; t += 256) sp += part_partials[t];
    for (int t = threadIdx.x; t < 64;   t += 256) sk += kld_partials[t];
    for (int t = threadIdx.x; t < 256;  t += 256) {
        s1  += pair_partials[2 * t];
        spp += pair_partials[2 * t + 1];
    }
    sp  = block_reduce_256(sp);
    sk  = block_reduce_256(sk);
    s1  = block_reduce_256(s1);
    spp = block_reduce_256(spp);
    if (threadIdx.x == 0) {
        const float part = sp - (float)N_POINTS;   // subtract diagonal
        out[0] = s1 + spp * logf(part) + 1e-7f * sk;
    }
}

// ---------------------------------------------------------------------------
// Launch
// ---------------------------------------------------------------------------
extern "C" void kernel_launch(void* const* d_in, const int* in_sizes, int n_in,
                              void* d_out, int out_size, void* d_ws, size_t ws_size,
                              hipStream_t stream) {
    const float* pij = (const float*)d_in[0];
    const int*   ii  = (const int*)  d_in[1];
    const int*   jj  = (const int*)  d_in[2];
    const float* mu  = (const float*)d_in[3];
    const float* lv  = (const float*)d_in[4];
    const float* ef  = (const float*)d_in[5];
    const float* ei  = (const float*)d_in[6];
    const float* ej  = (const float*)d_in[7];

    char* ws = (char*)d_ws;
    float4* Ar            = (float4*)(ws);                      // 16384*16 = 256 KB
    float4* Bc            = (float4*)(ws + 262144);             // 256 KB
    float*  part_partials = (float*) (ws + 524288);             // 8192 * 4 = 32 KB
    float*  kld_partials  = (float*) (ws + 524288 + 32768);     // 64 * 4
    float*  pair_partials = (float*) (ws + 524288 + 32768 + 256); // 512 * 4

    vtsne_prep<<<N_POINTS / 256, 256, 0, stream>>>(mu, lv, ef, Ar, Bc, kld_partials);

    vtsne_part_tiles<<<dim3(1024, 8), 32, 0, stream>>>(
        (const float2*)Ar, (const float2*)Bc, part_partials);

    vtsne_pairs<<<N_PAIRS / 256, 256, 0, stream>>>(
        pij, ii, jj, mu, lv, ei, ej, pair_partials);

    vtsne_finalize<<<1, 256, 0, stream>>>(part_partials, kld_partials,
                                          pair_partials, (float*)d_out);
}